// GeneralLagrangeSolverSubset_9861244912185
// MI455X (gfx1250) — compile-verified
//
#include <hip/hip_runtime.h>
#include <hip/hip_bf16.h>
#include <math.h>

// ---------------------------------------------------------------------------
// GeneralLagrangeSolverSubset — MI455X (gfx1250, wave32)
//   * V_WMMA_F32_16X16X4_F32 for the (16 x K)@(K x 16) Jacobian reductions
//   * GLOBAL_LOAD_ASYNC_TO_LDS_B128 (ASYNCcnt) for A-tile staging
//   * fused last-block GN solve (deterministic fixed-order reductions)
//
// r(mu)  = b0 + M0*mu + 0.05*B2@tanh(xf0 + G*mu) - z_new      (16)
// J(mu)  = M0 + 0.05*(B2 .* w)@G,  w = 1 - tanh^2             (16x16)
// GN: solve (J^T J) d = J^T r, mu -= d (done-masked), 30 iters
// pos = grad_xi @ (mu*m0/dt); mom = grad_xi @ Ginv(vz - (G/m0)^T p)
// ---------------------------------------------------------------------------

#define DIM      600000      // N_SUB * 3  (divisible by 4 -> b128 tiles all-or-nothing)
#define KT       256         // k-slab per block
#define LDA      260         // padded LDS stride (1040B % 256B = 16 -> conflict-free A frags)
#define NBLK     ((DIM + KT - 1) / KT)   // 2344
#define PSTRIDE  288         // [0:16) vec, [16:272) 16x16, [272:288) s
#define MAXIT    30
#define TOLF     1e-8f

// state layout (floats) in workspace
#define ST_M0    0
#define ST_B0    256
#define ST_S     272
#define ST_LM    288
#define ST_MU    304
#define ST_DONE  320
#define ST_CNT   324         // slot reused as uint counter
#define ST_SIZE  336

typedef __attribute__((ext_vector_type(2))) float v2f;
typedef __attribute__((ext_vector_type(8))) float v8f;
typedef int v4i __attribute__((vector_size(16)));   // builtin's b128 pointee type

#define AS1 __attribute__((address_space(1)))
#define AS3 __attribute__((address_space(3)))

#if defined(__has_builtin)
#if __has_builtin(__builtin_amdgcn_wmma_f32_16x16x4_f32)
#define USE_WMMA 1
#endif
#if __has_builtin(__builtin_amdgcn_global_load_async_to_lds_b128)
#define HAVE_ASYNC 1
#endif
#if __has_builtin(__builtin_amdgcn_s_wait_asynccnt)
#define HAVE_WAIT_ASYNC_BUILTIN 1
#endif
#endif

__device__ __forceinline__ void wait_async0() {
#ifdef HAVE_ASYNC
#ifdef HAVE_WAIT_ASYNC_BUILTIN
  __builtin_amdgcn_s_wait_asynccnt(0);
#else
  asm volatile("s_wait_asynccnt 0" ::: "memory");
#endif
#endif
}

// Stage the 16 x KT A-tile (rows of a (16 x DIM) row-major matrix starting at
// column kbase) into ldsA[16][LDA].  Async DMA path bypasses VGPRs; overlaps
// with the caller's pointwise phase until wait_async0().
__device__ __forceinline__ void stage_A_tile(const float* __restrict__ src,
                                             float* ldsA, int kbase, int tid) {
#ifdef HAVE_ASYNC
  #pragma unroll
  for (int p = 0; p < 4; ++p) {                  // 1024 b128 segments / 256 threads
    const int task = p * 256 + tid;
    const int i = task >> 6;                     // row 0..15
    const int kk = (task & 63) << 2;             // col 0,4,...,252
    if (kbase + kk + 3 < DIM) {                  // whole b128 in range
      __builtin_amdgcn_global_load_async_to_lds_b128(
          (AS1 v4i*)(void*)(src + (size_t)i * DIM + kbase + kk),
          (AS3 v4i*)(void*)(ldsA + i * LDA + kk), 0, 0);
    }
  }
  if (kbase + KT > DIM) {                        // tail block: zero OOB region only
    const int rem = DIM - kbase;                 // multiple of 4; disjoint from async
    for (int idx = tid; idx < 16 * KT; idx += 256) {
      const int i = idx >> 8, kk = idx & (KT - 1);
      if (kk >= rem) ldsA[i * LDA + kk] = 0.f;
    }
  }
#else
  #pragma unroll
  for (int i = 0; i < 16; ++i) {
    const int k = kbase + tid;
    ldsA[i * LDA + tid] = (k < DIM) ? src[(size_t)i * DIM + k] : 0.f;
  }
#endif
}

// (16 x KT) @ (KT x 16) via V_WMMA_F32_16X16X4_F32, fully unrolled (8 WMMAs/wave).
// ldsA: [16][LDA], ldsB: [KT][16], ldsC: [8][256] per-wave 16x16 partials.
// A-frag (ISA 7.12.2, 32-bit A 16x4): lanes0-15 M=lane; VGPR v holds K=2*half+v.
__device__ __forceinline__ void tile_matmul_16xK(const float* ldsA, const float* ldsB,
                                                 float* ldsC, int tid) {
#ifdef USE_WMMA
  const int wave = tid >> 5;
  const int lane = tid & 31;
  const int half = lane >> 4;
  const int m    = lane & 15;
  v8f acc = {};
  #pragma unroll
  for (int u = 0; u < KT / 32; ++u) {            // constant trip count -> unrolls
    const int kl = (wave + u * 8) << 2;          // wave-uniform: EXEC all-ones
    v2f a, b;
    a.x = ldsA[m * LDA + kl + 2 * half + 0];
    a.y = ldsA[m * LDA + kl + 2 * half + 1];
    b.x = ldsB[(kl + 2 * half + 0) * 16 + m];
    b.y = ldsB[(kl + 2 * half + 1) * 16 + m];
    acc = __builtin_amdgcn_wmma_f32_16x16x4_f32(false, a, false, b, (short)0, acc,
                                                false, false);
  }
  #pragma unroll
  for (int v = 0; v < 8; ++v)
    ldsC[wave * 256 + (v + 8 * half) * 16 + m] = acc[v];
#else
  const int i = tid >> 4, j = tid & 15;
  float s = 0.f;
  for (int k = 0; k < KT; ++k) s += ldsA[i * LDA + k] * ldsB[k * 16 + j];
  ldsC[tid] = s;
  #pragma unroll
  for (int w = 1; w < 8; ++w) ldsC[w * 256 + tid] = 0.f;
#endif
}

// --------------------------- setup pass (once) -----------------------------
__global__ __launch_bounds__(256) void p1_kernel(
    const float* __restrict__ x_nt, const float* __restrict__ mass,
    const float* __restrict__ gx,   const float* __restrict__ pt,
    const float* __restrict__ Bm,   const int* __restrict__ sidx,
    float* __restrict__ xf0_out,    float* __restrict__ partials) {
  __shared__ float ldsA[16 * LDA];
  __shared__ float ldsB[KT * 16];
  __shared__ float tb_xf[KT];
  __shared__ float tb_pt[KT];
  __shared__ float ldsC[8 * 256];

  const int tid = threadIdx.x;
  const int kbase = blockIdx.x * KT;
  const float m0 = mass[0];

  stage_A_tile(Bm, ldsA, kbase, tid);            // async DMA in flight

  const int k = kbase + tid;
  if (k < DIM) {
    const int s3 = k / 3, c = k - 3 * s3;
    const int atom = sidx[s3];
    const float xf = x_nt[atom * 3 + c];
    const float p  = pt[atom * 3 + c];
    const float invm = m0 / mass[k];
    xf0_out[k] = xf;
    tb_xf[tid] = xf;
    tb_pt[tid] = p / m0;                         // grad_xi/mass = G/m0
    const float* row = gx + (size_t)k * 16;
    #pragma unroll
    for (int j = 0; j < 16; ++j) ldsB[tid * 16 + j] = row[j] * invm;
  } else {
    tb_xf[tid] = 0.f; tb_pt[tid] = 0.f;
    #pragma unroll
    for (int j = 0; j < 16; ++j) ldsB[tid * 16 + j] = 0.f;
  }
  wait_async0();
  __syncthreads();

  tile_matmul_16xK(ldsA, ldsB, ldsC, tid);       // M0 partial
  __syncthreads();

  float* prec = partials + (size_t)blockIdx.x * PSTRIDE;
  { float s = 0.f;
    #pragma unroll
    for (int w = 0; w < 8; ++w) s += ldsC[w * 256 + tid];
    prec[16 + tid] = s; }
  if (tid < 16) {
    float b0 = 0.f, sj = 0.f;
    for (int kk = 0; kk < KT; ++kk) {
      b0 += ldsA[tid * LDA + kk] * tb_xf[kk];
      sj += ldsB[kk * 16 + tid] * tb_pt[kk];
    }
    prec[tid] = b0;
    prec[272 + tid] = sj;
  }
}

// ------------------- reduce setup partials, init solver state --------------
__global__ __launch_bounds__(288) void p2_kernel(
    const float* __restrict__ partials, const float* __restrict__ vz,
    const float* __restrict__ ginv, float* __restrict__ state) {
  __shared__ float sums[PSTRIDE];
  const int t = threadIdx.x;
  float s = 0.f;
  for (int b = 0; b < NBLK; ++b) s += partials[(size_t)b * PSTRIDE + t];
  sums[t] = s;
  __syncthreads();
  if (t < 256) state[ST_M0 + t] = sums[16 + t];
  if (t < 16) {
    state[ST_B0 + t] = sums[t];
    state[ST_S + t]  = sums[272 + t];
    float acc = 0.f;                             // lam_mom = Ginv @ (vz - s)
    #pragma unroll
    for (int i = 0; i < 16; ++i) acc += ginv[t * 16 + i] * (vz[i] - sums[272 + i]);
    state[ST_LM + t] = acc;
    state[ST_MU + t] = 0.f;
  }
  if (t == 0) {
    state[ST_DONE] = 0.f;
    *(unsigned*)(state + ST_CNT) = 0u;           // last-block election counter
  }
}

// ---------------- GN iteration: pass + fused last-block solve --------------
__global__ __launch_bounds__(256) void iter_kernel(
    const float* __restrict__ mass, const float* __restrict__ gx,
    const float* __restrict__ B2,   const float* __restrict__ xf0,
    const float* __restrict__ z_new, float* __restrict__ state,
    float* __restrict__ partials,   unsigned* __restrict__ cnt) {
  __shared__ float ldsA[16 * LDA];
  __shared__ float ldsB[KT * 16];
  __shared__ float tb_t[KT];
  __shared__ float ldsC[8 * 256];
  __shared__ float lmu[16];
  __shared__ float red[272];
  __shared__ float Jm[256];
  __shared__ float Nm[256];
  __shared__ float rv[16], gv[16];
  __shared__ float rn2;
  __shared__ int isLast;

  const int tid = threadIdx.x;
  const int kbase = blockIdx.x * KT;
  const float m0 = mass[0];

  if (tid < 16) lmu[tid] = state[ST_MU + tid];
  stage_A_tile(B2, ldsA, kbase, tid);            // async DMA in flight
  __syncthreads();                               // lmu visible; DMA still running

  const int k = kbase + tid;
  float tval = 0.f;
  if (k < DIM) {
    const float invm = m0 / mass[k];
    const float* row = gx + (size_t)k * 16;
    float xf = xf0[k];
    float grow[16];
    #pragma unroll
    for (int j = 0; j < 16; ++j) { grow[j] = row[j] * invm; xf += grow[j] * lmu[j]; }
    tval = tanhf(xf);
    const float w = 1.f - tval * tval;
    #pragma unroll
    for (int j = 0; j < 16; ++j) ldsB[tid * 16 + j] = w * grow[j];
  } else {
    #pragma unroll
    for (int j = 0; j < 16; ++j) ldsB[tid * 16 + j] = 0.f;
  }
  tb_t[tid] = tval;
  wait_async0();
  __syncthreads();

  tile_matmul_16xK(ldsA, ldsB, ldsC, tid);       // J2 partial
  __syncthreads();

  float* prec = partials + (size_t)blockIdx.x * PSTRIDE;
  { float s = 0.f;
    #pragma unroll
    for (int w = 0; w < 8; ++w) s += ldsC[w * 256 + tid];
    prec[16 + tid] = s; }
  if (tid < 16) {
    float r2 = 0.f;
    for (int kk = 0; kk < KT; ++kk) r2 += ldsA[tid * LDA + kk] * tb_t[kk];
    prec[tid] = r2;
  }
  __syncthreads();                               // all block stores before release RMW

  if (tid == 0) {
    const unsigned old = __hip_atomic_fetch_add(cnt, 1u, __ATOMIC_ACQ_REL,
                                                __HIP_MEMORY_SCOPE_AGENT);
    isLast = (old == (unsigned)(NBLK - 1));
  }
  __syncthreads();
  if (!isLast) return;

  // ---- elected block: deterministic fixed-order reduction + 16x16 GN solve
  { float s = 0.f;
    for (int b = 0; b < NBLK; ++b) s += partials[(size_t)b * PSTRIDE + tid];
    red[tid] = s; }
  if (tid < 16) {
    float s = 0.f;
    for (int b = 0; b < NBLK; ++b) s += partials[(size_t)b * PSTRIDE + 256 + tid];
    red[256 + tid] = s;
  }
  __syncthreads();

  Jm[tid] = state[ST_M0 + tid] + 0.05f * red[16 + tid];
  if (tid < 16) {
    float acc = state[ST_B0 + tid] + 0.05f * red[tid] - z_new[tid];
    #pragma unroll
    for (int j = 0; j < 16; ++j) acc += state[ST_M0 + tid * 16 + j] * lmu[j];
    rv[tid] = acc;
  }
  __syncthreads();

  { const int i = tid >> 4, j = tid & 15;
    float acc = 0.f;
    #pragma unroll
    for (int kk = 0; kk < 16; ++kk) acc += Jm[kk * 16 + i] * Jm[kk * 16 + j];
    Nm[tid] = acc; }                             // J^T J
  if (tid < 16) {
    float acc = 0.f;
    #pragma unroll
    for (int kk = 0; kk < 16; ++kk) acc += Jm[kk * 16 + tid] * rv[kk];
    gv[tid] = acc;                               // J^T r
  }
  if (tid == 0) {
    float a = 0.f;
    #pragma unroll
    for (int kk = 0; kk < 16; ++kk) a += rv[kk] * rv[kk];
    rn2 = a;
  }
  __syncthreads();

  if (tid == 0) {                                // pivoted Gaussian elim, serial
    float delta[16];
    for (int col = 0; col < 16; ++col) {
      int piv = col; float best = fabsf(Nm[col * 16 + col]);
      for (int r2 = col + 1; r2 < 16; ++r2) {
        const float v = fabsf(Nm[r2 * 16 + col]);
        if (v > best) { best = v; piv = r2; }
      }
      if (piv != col) {
        for (int cc = col; cc < 16; ++cc) {
          const float tmp = Nm[col * 16 + cc];
          Nm[col * 16 + cc] = Nm[piv * 16 + cc];
          Nm[piv * 16 + cc] = tmp;
        }
        const float tg = gv[col]; gv[col] = gv[piv]; gv[piv] = tg;
      }
      const float inv = 1.f / Nm[col * 16 + col];
      for (int r2 = col + 1; r2 < 16; ++r2) {
        const float f = Nm[r2 * 16 + col] * inv;
        for (int cc = col; cc < 16; ++cc) Nm[r2 * 16 + cc] -= f * Nm[col * 16 + cc];
        gv[r2] -= f * gv[col];
      }
    }
    for (int r2 = 15; r2 >= 0; --r2) {
      float acc = gv[r2];
      for (int cc = r2 + 1; cc < 16; ++cc) acc -= Nm[r2 * 16 + cc] * delta[cc];
      delta[r2] = acc / Nm[r2 * 16 + r2];
    }
    const float done = state[ST_DONE];
    if (done == 0.f)
      for (int j = 0; j < 16; ++j) state[ST_MU + j] = lmu[j] - delta[j];
    if (sqrtf(rn2) < TOLF) state[ST_DONE] = 1.f;
    __hip_atomic_store(cnt, 0u, __ATOMIC_RELAXED, __HIP_MEMORY_SCOPE_AGENT);
  }
}

// --------------------------- final outputs ---------------------------------
__global__ __launch_bounds__(256) void final_kernel(
    const float* __restrict__ gx, const float* __restrict__ mass,
    const float* __restrict__ dtP, const float* __restrict__ state,
    float* __restrict__ out) {
  __shared__ float lam[16], lm[16];
  const int t = threadIdx.x;
  if (t < 16) {
    lam[t] = state[ST_MU + t] * (mass[0] / dtP[0]);
    lm[t]  = state[ST_LM + t];
  }
  __syncthreads();
  const int k = blockIdx.x * 256 + t;
  if (k < DIM) {
    const float* row = gx + (size_t)k * 16;
    float p = 0.f, m = 0.f;
    #pragma unroll
    for (int j = 0; j < 16; ++j) { const float v = row[j]; p += v * lam[j]; m += v * lm[j]; }
    out[k] = p;
    out[DIM + k] = m;
  }
}

// ---------------------------------------------------------------------------
extern "C" void kernel_launch(void* const* d_in, const int* in_sizes, int n_in,
                              void* d_out, int out_size, void* d_ws, size_t ws_size,
                              hipStream_t stream) {
  (void)in_sizes; (void)n_in; (void)out_size; (void)ws_size;
  // setup_inputs order: x, x_newtilde, z_new, dt, mass_subset, grad_xi_subset,
  //                     ptilde, vz, grad_xi_gram_inv, B, B2, subset_indices
  const float* x_nt  = (const float*)d_in[1];
  const float* z_new = (const float*)d_in[2];
  const float* dt    = (const float*)d_in[3];
  const float* mass  = (const float*)d_in[4];
  const float* gx    = (const float*)d_in[5];
  const float* pt    = (const float*)d_in[6];
  const float* vz    = (const float*)d_in[7];
  const float* ginv  = (const float*)d_in[8];
  const float* Bm    = (const float*)d_in[9];
  const float* B2    = (const float*)d_in[10];
  const int*   sidx  = (const int*)d_in[11];

  float* ws       = (float*)d_ws;
  float* xf0      = ws;                                   // DIM
  float* partials = ws + DIM;                             // NBLK*PSTRIDE
  float* state    = partials + (size_t)NBLK * PSTRIDE;    // ST_SIZE
  unsigned* cnt   = (unsigned*)(state + ST_CNT);

  p1_kernel<<<NBLK, 256, 0, stream>>>(x_nt, mass, gx, pt, Bm, sidx, xf0, partials);
  p2_kernel<<<1, 288, 0, stream>>>(partials, vz, ginv, state);
  for (int it = 0; it < MAXIT; ++it)
    iter_kernel<<<NBLK, 256, 0, stream>>>(mass, gx, B2, xf0, z_new, state,
                                          partials, cnt);
  final_kernel<<<NBLK, 256, 0, stream>>>(gx, mass, dt, state, (float*)d_out);
}